// LSEPLoss2_85830626443704
// MI455X (gfx1250) — compile-verified
//
#include <hip/hip_runtime.h>
#include <hip/hip_bf16.h>
#include <stdint.h>

#define NUM_CLASSES 340
#define CHUNKS 85          // 340 floats = 85 x 16B
#define WAVES_PER_BLOCK 8
#define BLOCK_THREADS 256

// Issue one row (1360 B) as 85 async b128 global->LDS transfers (3 instructions).
__device__ __forceinline__ void issue_row_async(const float* __restrict__ rowp,
                                                uint32_t ldsAddr, int lane) {
    uint64_t ga = (uint64_t)(uintptr_t)rowp + (uint64_t)(lane * 16);
    uint32_t la = ldsAddr + (uint32_t)(lane * 16);
    // chunks [0,32): bytes lane*16
    asm volatile("global_load_async_to_lds_b128 %0, %1, off"
                 :: "v"(la), "v"(ga) : "memory");
    // chunks [32,64): +512 B
    asm volatile("global_load_async_to_lds_b128 %0, %1, off offset:512"
                 :: "v"(la), "v"(ga) : "memory");
    // chunks [64,85): +1024 B, lanes 0..20 only
    if (lane < (CHUNKS - 64)) {
        asm volatile("global_load_async_to_lds_b128 %0, %1, off offset:1024"
                     :: "v"(la), "v"(ga) : "memory");
    }
}

__global__ __launch_bounds__(BLOCK_THREADS)
void lsep_partial_kernel(const float* __restrict__ inp,
                         const int* __restrict__ target,
                         double* __restrict__ blockPartials,
                         int B) {
    __shared__ __align__(16) float ldsBuf[WAVES_PER_BLOCK][2][NUM_CLASSES];
    __shared__ double redArr[WAVES_PER_BLOCK];

    const int lane = threadIdx.x & 31;
    const int wave = threadIdx.x >> 5;
    const int gw     = blockIdx.x * WAVES_PER_BLOCK + wave;   // global wave id
    const int stride = gridDim.x * WAVES_PER_BLOCK;           // total waves

    uint32_t ldsBase[2];
    ldsBase[0] = (uint32_t)(uintptr_t)&ldsBuf[wave][0][0];
    ldsBase[1] = (uint32_t)(uintptr_t)&ldsBuf[wave][1][0];

    double wacc = 0.0;
    int row = gw;
    int cb = 0;

    if (row < B) {
        issue_row_async(inp + (size_t)row * NUM_CLASSES, ldsBase[0], lane);
    }

    for (; row < B; row += stride) {
        const int nxt = row + stride;
        const bool hasNext = (nxt < B);

        if (hasNext) {
            issue_row_async(inp + (size_t)nxt * NUM_CLASSES, ldsBase[cb ^ 1], lane);
            // <=3 outstanding => the 3 transfers of the current buffer are done
            asm volatile("s_wait_asynccnt 0x3" ::: "memory");
        } else {
            asm volatile("s_wait_asynccnt 0x0" ::: "memory");
        }

        const float* buf = &ldsBuf[wave][cb][0];
        float acc = 0.0f;
        #pragma unroll
        for (int j = lane; j < NUM_CLASSES; j += 32) {
            acc += expf(buf[j]);
        }
        // wave32 butterfly reduction: all lanes end with the row sum S
        #pragma unroll
        for (int m = 16; m >= 1; m >>= 1) {
            acc += __shfl_xor(acc, m, 32);
        }

        const int   y   = target[row];
        const float pos = buf[y];                 // LDS broadcast read
        wacc += (double)(acc * expf(-pos));       // exp(-pos) * S_i

        cb ^= 1;
    }

    if (lane == 0) redArr[wave] = wacc;
    __syncthreads();
    if (threadIdx.x == 0) {
        double s = 0.0;
        #pragma unroll
        for (int w = 0; w < WAVES_PER_BLOCK; ++w) s += redArr[w];
        blockPartials[blockIdx.x] = s;            // deterministic fixed order
    }
}

__global__ __launch_bounds__(BLOCK_THREADS)
void lsep_final_kernel(const double* __restrict__ partials, int n,
                       float* __restrict__ out, double subConst) {
    __shared__ double sm[BLOCK_THREADS];
    double s = 0.0;
    for (int i = threadIdx.x; i < n; i += BLOCK_THREADS) s += partials[i];
    sm[threadIdx.x] = s;
    __syncthreads();
    for (int off = BLOCK_THREADS / 2; off > 0; off >>= 1) {
        if (threadIdx.x < off) sm[threadIdx.x] += sm[threadIdx.x + off];
        __syncthreads();
    }
    if (threadIdx.x == 0) {
        // pair_sum = sum_i exp(-pos_i)*S_i - B ; loss = log1p(pair_sum)
        out[0] = (float)log1p(sm[0] - subConst);
    }
}

extern "C" void kernel_launch(void* const* d_in, const int* in_sizes, int n_in,
                              void* d_out, int out_size, void* d_ws, size_t ws_size,
                              hipStream_t stream) {
    (void)n_in; (void)out_size; (void)ws_size;
    const float* inp    = (const float*)d_in[0];
    const int*   target = (const int*)d_in[1];
    float*       out    = (float*)d_out;
    const int B = in_sizes[1];

    const int NB = 2048;  // 16384 waves -> 4 rows/wave at B=65536; 16 KB of ws
    double* partials = (double*)d_ws;

    lsep_partial_kernel<<<NB, BLOCK_THREADS, 0, stream>>>(inp, target, partials, B);
    lsep_final_kernel<<<1, BLOCK_THREADS, 0, stream>>>(partials, NB, out, (double)B);
}